// Matcher_29918742184163
// MI455X (gfx1250) — compile-verified
//
#include <hip/hip_runtime.h>
#include <hip/hip_bf16.h>
#include <stdint.h>

#define NUM_O  20
#define QP     8192
#define BS     16
#define TQ     512          // queries per tile
#define NT     512          // threads per block (16 waves of 32)
#define NTILES (QP / TQ)    // 16
#define NWAVES (NT / 32)    // 16

// LDS byte offset of a shared-memory generic pointer: per CDNA5 aperture
// rules the LDS offset is the low 32 bits of the flat address.
__device__ __forceinline__ uint32_t lds_lo32(const void* p) {
  return (uint32_t)(uintptr_t)p;
}

// gfx1250 async copy: 16B global -> LDS, tracked by ASYNCcnt.
__device__ __forceinline__ void async_copy_b128(uint32_t lds_byte, uint64_t gaddr) {
  asm volatile("global_load_async_to_lds_b128 %0, %1, off"
               :: "v"(lds_byte), "v"(gaddr)
               : "memory");
}

__device__ __forceinline__ void async_wait_all() {
  asm volatile("s_wait_asynccnt 0" ::: "memory");
}

__global__ __launch_bounds__(NT) void matcher_kernel(
    const float* __restrict__ pred_logits,   // (bs, O*QP, 1)
    const float* __restrict__ anchors,       // (O*QP, 6)
    const float* __restrict__ target_boxes,  // (bs, O, 6) cxcyczwhd
    const int*   __restrict__ target_present,// (bs, O)
    int*   __restrict__ matches_out,         // (bs, O, QP) int32
    float* __restrict__ soft_out)            // (bs, O, QP) float32
{
  __shared__ alignas(16) float s_cg[QP];          // 32 KB: cost_giou cache
  __shared__ alignas(16) float s_anc[2][TQ * 6];  // 24 KB: double-buffered anchors
  __shared__ alignas(16) float s_log[2][TQ];      //  4 KB: double-buffered logits
  __shared__ float s_rbc[NWAVES];
  __shared__ int   s_rbq[NWAVES];
  __shared__ float s_rmn[NWAVES];
  __shared__ float s_rmx[NWAVES];

  const int bo  = blockIdx.x;
  const int b   = bo / NUM_O;
  const int o   = bo % NUM_O;
  const int tid = threadIdx.x;

  // Target box (uniform per block): convert to corners once.
  const float* tb = target_boxes + (size_t)bo * 6;
  const float tcx = tb[0], tcy = tb[1], tcz = tb[2];
  const float tsx = tb[3], tsy = tb[4], tsz = tb[5];
  const float bx0 = tcx - 0.5f * tsx, by0 = tcy - 0.5f * tsy, bz0 = tcz - 0.5f * tsz;
  const float bx1 = tcx + 0.5f * tsx, by1 = tcy + 0.5f * tsy, bz1 = tcz + 0.5f * tsz;
  const float vol_b = tsx * tsy * tsz;
  const int present = target_present[bo];

  const uint64_t anc_base = (uint64_t)(uintptr_t)(anchors + (size_t)o * QP * 6);
  const uint64_t log_base = (uint64_t)(uintptr_t)(pred_logits +
                                                  (size_t)b * (NUM_O * QP) +
                                                  (size_t)o * QP);

  auto stage = [&](int tile, int buf) {
    // anchors: TQ*24 = 12288 bytes = 768 x b128
    uint64_t ga = anc_base + (uint64_t)tile * (TQ * 24u);
    uint32_t la = lds_lo32(&s_anc[buf][0]);
    async_copy_b128(la + (uint32_t)tid * 16u, ga + (uint64_t)tid * 16u);
    if (tid < (TQ * 24 / 16) - NT)  // remaining 256 chunks
      async_copy_b128(la + (uint32_t)(NT + tid) * 16u, ga + (uint64_t)(NT + tid) * 16u);
    // logits: TQ*4 = 2048 bytes = 128 x b128
    uint64_t gl = log_base + (uint64_t)tile * (TQ * 4u);
    uint32_t ll = lds_lo32(&s_log[buf][0]);
    if (tid < (TQ * 4 / 16))
      async_copy_b128(ll + (uint32_t)tid * 16u, gl + (uint64_t)tid * 16u);
  };

  // Prologue: stage tile 0 and wait.
  stage(0, 0);
  async_wait_all();
  __syncthreads();

  float bestC = INFINITY;
  int   bestQ = 0;
  float cg_mn = INFINITY;
  float cg_mx = -INFINITY;

  for (int t = 0; t < NTILES; ++t) {
    const int buf = t & 1;
    if (t + 1 < NTILES) stage(t + 1, buf ^ 1);  // prefetch next tile (async)

    // One query per thread per tile; q strictly increases with t.
    const int q  = t * TQ + tid;
    const float* a = &s_anc[buf][tid * 6];
    const float ax = a[0], ay = a[1], az = a[2];
    const float aw = a[3], ah = a[4], ad = a[5];
    const float lg = s_log[buf][tid];

    // cost_bbox on raw anchors
    const float cb = fabsf(ax - tcx) + fabsf(ay - tcy) + fabsf(az - tcz) +
                     fabsf(aw - tsx) + fabsf(ah - tsy) + fabsf(ad - tsz);

    // giou on clip(anchor, 0)
    const float cx = fmaxf(ax, 0.f), cy = fmaxf(ay, 0.f), cz = fmaxf(az, 0.f);
    const float cw = fmaxf(aw, 0.f), ch = fmaxf(ah, 0.f), cd = fmaxf(ad, 0.f);
    const float ax0 = cx - 0.5f * cw, ay0 = cy - 0.5f * ch, az0 = cz - 0.5f * cd;
    const float ax1 = cx + 0.5f * cw, ay1 = cy + 0.5f * ch, az1 = cz + 0.5f * cd;
    const float vol_a = cw * ch * cd;

    const float ix = fmaxf(fminf(ax1, bx1) - fmaxf(ax0, bx0), 0.f);
    const float iy = fmaxf(fminf(ay1, by1) - fmaxf(ay0, by0), 0.f);
    const float iz = fmaxf(fminf(az1, bz1) - fmaxf(az0, bz0), 0.f);
    const float inter = ix * iy * iz;
    const float uni   = vol_a + vol_b - inter;
    const float iou   = inter / uni;

    const float ex = fmaxf(fmaxf(ax1, bx1) - fminf(ax0, bx0), 0.f);
    const float ey = fmaxf(fmaxf(ay1, by1) - fminf(ay0, by0), 0.f);
    const float ez = fmaxf(fmaxf(az1, bz1) - fminf(az0, bz0), 0.f);
    const float volc = ex * ey * ez;

    const float cg  = -(iou - (volc - uni) / volc);   // cost_giou
    const float sig = 1.f / (1.f + __expf(-lg));
    const float C   = 5.f * cb - 2.f * sig + 2.f * cg;

    s_cg[q] = cg;
    if (C < bestC) { bestC = C; bestQ = q; }  // strict < keeps first occurrence
    cg_mn = fminf(cg_mn, cg);
    cg_mx = fmaxf(cg_mx, cg);

    if (t + 1 < NTILES) async_wait_all();  // next tile landed during compute
    __syncthreads();
  }

  // Wave32 reduction: lexicographic (C, q) argmin + cg min/max.
  for (int off = 16; off; off >>= 1) {
    const float oc = __shfl_xor(bestC, off, 32);
    const int   oq = __shfl_xor(bestQ, off, 32);
    if (oc < bestC || (oc == bestC && oq < bestQ)) { bestC = oc; bestQ = oq; }
    const float omn = __shfl_xor(cg_mn, off, 32);
    const float omx = __shfl_xor(cg_mx, off, 32);
    cg_mn = fminf(cg_mn, omn);
    cg_mx = fmaxf(cg_mx, omx);
  }
  const int wave = tid >> 5;
  if ((tid & 31) == 0) {
    s_rbc[wave] = bestC; s_rbq[wave] = bestQ;
    s_rmn[wave] = cg_mn; s_rmx[wave] = cg_mx;
  }
  __syncthreads();

  float fbc = s_rbc[0]; int fbq = s_rbq[0];
  float fmn = s_rmn[0]; float fmx = s_rmx[0];
  #pragma unroll
  for (int w = 1; w < NWAVES; ++w) {
    const float oc = s_rbc[w]; const int oq = s_rbq[w];
    if (oc < fbc || (oc == fbc && oq < fbq)) { fbc = oc; fbq = oq; }
    fmn = fminf(fmn, s_rmn[w]);
    fmx = fmaxf(fmx, s_rmx[w]);
  }

  // Pass 2: emit matches + soft_labels (coalesced).
  const float inv = 1.f / (fmn - fmx);
  const size_t obase = (size_t)bo * QP;
  for (int q = tid; q < QP; q += NT) {
    float sl; int m;
    if (present) {
      sl = fmaxf((s_cg[q] - fmx) * inv, 0.f);
      m  = (q == fbq) ? 1 : 0;
    } else {
      sl = -1.f;
      m  = 0;
    }
    soft_out[obase + q]    = sl;
    matches_out[obase + q] = m;
  }
}

extern "C" void kernel_launch(void* const* d_in, const int* in_sizes, int n_in,
                              void* d_out, int out_size, void* d_ws, size_t ws_size,
                              hipStream_t stream) {
  (void)in_sizes; (void)n_in; (void)out_size; (void)d_ws; (void)ws_size;
  const float* pred_logits    = (const float*)d_in[0];
  // d_in[1] = pred_boxes: unused by the reference computation -> never read (saves 39 MB of traffic)
  const float* anchors        = (const float*)d_in[2];
  const float* target_boxes   = (const float*)d_in[3];
  const int*   target_present = (const int*)d_in[4];
  // d_in[5] = num_top_queries == 1 (top-1 argmin path)

  int*   matches_out = (int*)d_out;
  float* soft_out    = (float*)d_out + (size_t)BS * NUM_O * QP;

  matcher_kernel<<<BS * NUM_O, NT, 0, stream>>>(
      pred_logits, anchors, target_boxes, target_present, matches_out, soft_out);
}